// Net_2_79285096284771
// MI455X (gfx1250) — compile-verified
//
#include <hip/hip_runtime.h>

#define EMB_DIM 128
#define OUT_DIM 5
#define BAGS_PER_BLOCK 16
#define BLOCK_THREADS 256   // 8 waves (wave32)
#define BAGS_PER_WAVE 2
#define POOL_STRIDE 132     // 128 + 4 pad -> conflict-free LDS (bank = 4*m + k)

typedef __attribute__((ext_vector_type(2))) float v2f;
typedef __attribute__((ext_vector_type(8))) float v8f;

__launch_bounds__(BLOCK_THREADS)
__global__ void embbag_linear_kernel(const float* __restrict__ emb,
                                     const float* __restrict__ lin_w,
                                     const float* __restrict__ lin_b,
                                     const int*   __restrict__ sf,
                                     const int*   __restrict__ offs,
                                     float* __restrict__ out,
                                     int B) {
    __shared__ __align__(16) float s_pool[BAGS_PER_BLOCK][POOL_STRIDE]; // pooled 16x128 f32 tile
    __shared__ __align__(16) float s_w[16][POOL_STRIDE];                // zero-padded W^T tile (16x128)

    const int tid  = threadIdx.x;
    const int wave = tid >> 5;
    const int lane = tid & 31;
    const int bag0 = blockIdx.x * BAGS_PER_BLOCK;

    // Cooperative fill of the zero-padded weight tile: s_w[n][k] = (n<5) ? lin_w[n][k] : 0
    for (int t = tid; t < 16 * EMB_DIM; t += BLOCK_THREADS) {
        const int n = t >> 7;
        const int k = t & (EMB_DIM - 1);
        s_w[n][k] = (n < OUT_DIM) ? lin_w[n * EMB_DIM + k] : 0.0f;
    }

    // -------- Phase 1: mean-pool BAGS_PER_WAVE bags per wave --------
    // Lane l owns floats [4l, 4l+4) of the 128-dim row: one global_load_b128 per index.
    const float* base = emb + (size_t)lane * 4;
    for (int jj = 0; jj < BAGS_PER_WAVE; ++jj) {
        const int j   = wave * BAGS_PER_WAVE + jj;
        const int bag = bag0 + j;
        int start = 0, end = 0;
        if (bag < B) { start = offs[bag]; end = offs[bag + 1]; }

        float ax = 0.f, ay = 0.f, az = 0.f, aw = 0.f;
        int i = start;
        // 8 B128 loads in flight per wave for memory-level parallelism
        for (; i + 8 <= end; i += 8) {
            const int i0 = sf[i + 0], i1 = sf[i + 1], i2 = sf[i + 2], i3 = sf[i + 3];
            const int i4 = sf[i + 4], i5 = sf[i + 5], i6 = sf[i + 6], i7 = sf[i + 7];
            const float4 v0 = *(const float4*)(base + (size_t)i0 * EMB_DIM);
            const float4 v1 = *(const float4*)(base + (size_t)i1 * EMB_DIM);
            const float4 v2 = *(const float4*)(base + (size_t)i2 * EMB_DIM);
            const float4 v3 = *(const float4*)(base + (size_t)i3 * EMB_DIM);
            const float4 v4 = *(const float4*)(base + (size_t)i4 * EMB_DIM);
            const float4 v5 = *(const float4*)(base + (size_t)i5 * EMB_DIM);
            const float4 v6 = *(const float4*)(base + (size_t)i6 * EMB_DIM);
            const float4 v7 = *(const float4*)(base + (size_t)i7 * EMB_DIM);
            ax += v0.x + v1.x + v2.x + v3.x + v4.x + v5.x + v6.x + v7.x;
            ay += v0.y + v1.y + v2.y + v3.y + v4.y + v5.y + v6.y + v7.y;
            az += v0.z + v1.z + v2.z + v3.z + v4.z + v5.z + v6.z + v7.z;
            aw += v0.w + v1.w + v2.w + v3.w + v4.w + v5.w + v6.w + v7.w;
        }
        for (; i < end; ++i) {
            const int i0 = sf[i];
            const float4 v0 = *(const float4*)(base + (size_t)i0 * EMB_DIM);
            ax += v0.x; ay += v0.y; az += v0.z; aw += v0.w;
        }
        const int cnt = end - start;
        const float inv = 1.0f / (float)(cnt > 1 ? cnt : 1);
        *(float4*)&s_pool[j][lane * 4] = make_float4(ax * inv, ay * inv, az * inv, aw * inv);
    }

    __syncthreads();

    // -------- Phase 2: [16 bags x 128] @ [128 x 16 (5 valid)] + bias via WMMA --------
    // Wave-uniform branch -> EXEC all ones inside (WMMA requirement).
    if (wave == 0) {
        const int half = lane >> 4;   // K sub-block selector (A/B), row-offset selector (C/D)
        const int mn   = lane & 15;   // A row m / B col n / D col n

        const float bias = (mn < OUT_DIM) ? lin_b[mn] : 0.0f;
        v8f c;
#pragma unroll
        for (int r = 0; r < 8; ++r) c[r] = bias;

        // Chain 32 x V_WMMA_F32_16X16X4_F32 over K = 128.
        // A (16x4 f32): lane m holds K = 2*half, 2*half+1  (ISA 7.12.2)
        // B (4x16 f32): lane n holds K = 2*half, 2*half+1
#pragma unroll 4
        for (int k0 = 0; k0 < EMB_DIM; k0 += 4) {
            const int kk = k0 + half * 2;
            v2f a, b;
            a.x = s_pool[mn][kk];
            a.y = s_pool[mn][kk + 1];
            b.x = s_w[mn][kk];
            b.y = s_w[mn][kk + 1];
            c = __builtin_amdgcn_wmma_f32_16x16x4_f32(false, a, false, b,
                                                      (short)0, c, false, false);
        }

        // D layout: VGPR r -> row M = r + 8*half, col N = mn (only N < 5 valid)
#pragma unroll
        for (int r = 0; r < 8; ++r) {
            const int bag = bag0 + r + half * 8;
            if (mn < OUT_DIM && bag < B)
                out[bag * OUT_DIM + mn] = c[r];
        }
    }
}

extern "C" void kernel_launch(void* const* d_in, const int* in_sizes, int n_in,
                              void* d_out, int out_size, void* d_ws, size_t ws_size,
                              hipStream_t stream) {
    const float* emb   = (const float*)d_in[0];
    const float* lin_w = (const float*)d_in[1];
    const float* lin_b = (const float*)d_in[2];
    const int*   sf    = (const int*)d_in[3];
    const int*   offs  = (const int*)d_in[4];
    float* out = (float*)d_out;

    const int B = in_sizes[4] - 1;  // offsets has B+1 entries
    const int grid = (B + BAGS_PER_BLOCK - 1) / BAGS_PER_BLOCK;
    embbag_linear_kernel<<<grid, BLOCK_THREADS, 0, stream>>>(emb, lin_w, lin_b, sf, offs, out, B);
}